// CellGNN_79113297592869
// MI455X (gfx1250) — compile-verified
//
#include <hip/hip_runtime.h>
#include <hip/hip_bf16.h>
#include <math.h>

typedef __attribute__((ext_vector_type(2))) float v2f;
typedef __attribute__((ext_vector_type(8))) float v8f;

#define BB    8
#define NN_N  32
#define CDIM  32
#define MM    64
#define HDIM  64
#define KNN   16
#define NMPB  2048                 // nodes per batch (N*M)
#define NTOT  16384                // total nodes
#define NEDGE (NTOT * KNN)         // knn edges = 262144
#define NETOT (NEDGE + NTOT)       // + self loops = 278528
#define NEG_SLOPE 0.2f

// ---------- float <-> orderable uint (for atomic max on float) ----------
__device__ __forceinline__ unsigned fenc(float f) {
    unsigned u = __float_as_uint(f);
    return (u & 0x80000000u) ? ~u : (u | 0x80000000u);
}
__device__ __forceinline__ float fdec(unsigned u) {
    return (u & 0x80000000u) ? __uint_as_float(u & 0x7FFFFFFFu)
                             : __uint_as_float(~u);
}
// fenc(-inf) = ~0xFF800000 = 0x007FFFFF
#define ENC_NEG_INF 0x007FFFFFu

// ---------- 1) transpose x -> nodes [NTOT][32], plus squared norms ----------
__global__ __launch_bounds__(256) void prep_kernel(const float* __restrict__ x,
                                                   float* __restrict__ nodes,
                                                   float* __restrict__ sq) {
    int g = blockIdx.x * 256 + threadIdx.x;          // node id
    if (g >= NTOT) return;
    int b = g >> 11, rem = g & (NMPB - 1);
    int n = rem >> 6, m = rem & 63;
    const float* xp = x + ((size_t)(b * NN_N + n) * CDIM) * MM + m;  // + c*MM
    float s = 0.f;
    #pragma unroll
    for (int c = 0; c < CDIM; ++c) {
        float v = xp[c * MM];
        nodes[(size_t)g * CDIM + c] = v;
        s += v * v;
    }
    sq[g] = s;
}

// ---------- 2) kNN: WMMA distance tiles + in-LDS top-17 ----------
// grid = B * (2048/16) = 1024 blocks, 512 threads (16 waves)
// dyn LDS: dist[16][2048] + rowc[16][32] + rsq[16]
__global__ __launch_bounds__(512) void knn_kernel(const float* __restrict__ nodes,
                                                  const float* __restrict__ sq,
                                                  int* __restrict__ knn) {
    extern __shared__ float smem[];
    float* dist = smem;                    // 16*2048
    float* rowc = smem + 16 * NMPB;        // 16*32
    float* rsq  = rowc + 16 * CDIM;        // 16

    int b  = blockIdx.x >> 7;
    int rt = blockIdx.x & 127;
    int rbase = b * NMPB + rt * 16;        // global id of first row node
    int tid  = threadIdx.x;
    int wave = tid >> 5;
    int lane = tid & 31;

    for (int i = tid; i < 16 * CDIM; i += 512)
        rowc[i] = nodes[(size_t)(rbase + (i >> 5)) * CDIM + (i & 31)];
    if (tid < 16) rsq[tid] = sq[rbase + tid];
    __syncthreads();

    int m16    = lane & 15;
    int koff   = (lane >> 4) << 1;         // 0 for lanes 0-15, 2 for lanes 16-31
    int rowoff = (lane >> 4) << 3;         // rows 0-7 / 8-15
    const float* cnodes = nodes + (size_t)b * NMPB * CDIM;
    const float* sqb    = sq + b * NMPB;

    // Hoist loop-invariant A operands (row coords) and row norms to registers:
    // keeps the hot loop free of LDS loads so WMMAs only wait on B-operand VMEM.
    v2f aop[8];
    #pragma unroll
    for (int s = 0; s < 8; ++s) {
        int k = 4 * s + koff;
        aop[s].x = rowc[m16 * CDIM + k];
        aop[s].y = rowc[m16 * CDIM + k + 1];
    }
    float rs[8];
    #pragma unroll
    for (int v = 0; v < 8; ++v) rs[v] = rsq[rowoff + v];

    // Phase 1: each wave computes column tiles ct = wave, wave+16, ...
    for (int ct = wave; ct < NMPB / 16; ct += 16) {
        int cbase = ct * 16;
        const float* cptr = cnodes + (size_t)(cbase + m16) * CDIM;
        if (ct + 16 < NMPB / 16)   // prefetch next tile's column coords
            __builtin_prefetch(cnodes + (size_t)(cbase + 256 + m16) * CDIM, 0, 1);
        float cs = sqb[cbase + m16];
        v8f acc = {};
        #pragma unroll
        for (int s = 0; s < 8; ++s) {
            int k = 4 * s + koff;
            v2f bm; bm.x = cptr[k]; bm.y = cptr[k + 1];
            acc = __builtin_amdgcn_wmma_f32_16x16x4_f32(
                false, aop[s], false, bm, (short)0, acc, false, false);
        }
        #pragma unroll
        for (int v = 0; v < 8; ++v)
            dist[(rowoff + v) * NMPB + cbase + m16] = rs[v] + cs - 2.0f * acc[v];
    }
    __syncthreads();

    // Phase 2: wave w selects 17 smallest for row w; drop the first (self)
    int r = wave;
    float* drow = dist + r * NMPB;
    int gsrc = rbase + r;
    for (int it = 0; it < KNN + 1; ++it) {
        float best = 3.0e38f; int bidx = -1;
        for (int c = lane; c < NMPB; c += 32) {
            float dv = drow[c];
            if (dv < best || (dv == best && (unsigned)c < (unsigned)bidx)) {
                best = dv; bidx = c;
            }
        }
        #pragma unroll
        for (int off = 16; off >= 1; off >>= 1) {
            float ov = __shfl_xor(best, off, 32);
            int   oi = __shfl_xor(bidx, off, 32);
            if (ov < best || (ov == best && (unsigned)oi < (unsigned)bidx)) {
                best = ov; bidx = oi;
            }
        }
        if (lane == 0) {
            drow[bidx] = 3.0e38f;                       // exclude from later rounds
            if (it > 0) knn[gsrc * KNN + (it - 1)] = b * NMPB + bidx;
        }
        __syncthreads();    // uniform across all 16 waves; makes LDS update visible
    }
}

// ---------- 3) features: H = X @ W^T via f32 WMMA (KDIM compile-time) ----------
// grid = NTOT/16, block = 128 (4 waves -> 4 output col tiles of 16)
template <int KDIM>
__global__ __launch_bounds__(128) void feat_kernel(const float* __restrict__ X,
                                                   const float* __restrict__ W,
                                                   float* __restrict__ Hout) {
    int rbase = blockIdx.x * 16;
    int wave = threadIdx.x >> 5, lane = threadIdx.x & 31;
    int m16 = lane & 15, koff = (lane >> 4) << 1;
    const float* xr = X + (size_t)(rbase + m16) * KDIM;
    const float* wr = W + (size_t)(wave * 16 + m16) * KDIM;
    v8f acc = {};
    #pragma unroll
    for (int k0 = 0; k0 < KDIM; k0 += 4) {
        int k = k0 + koff;
        v2f a;  a.x = xr[k]; a.y = xr[k + 1];
        v2f bm; bm.x = wr[k]; bm.y = wr[k + 1];
        acc = __builtin_amdgcn_wmma_f32_16x16x4_f32(
            false, a, false, bm, (short)0, acc, false, false);
    }
    int rowoff = (lane >> 4) << 3;
    #pragma unroll
    for (int v = 0; v < 8; ++v)
        Hout[(size_t)(rbase + v + rowoff) * HDIM + wave * 16 + m16] = acc[v];
}

// ---------- 4) per-node attention logits ----------
__global__ __launch_bounds__(256) void alpha_kernel(const float* __restrict__ Hm,
                                                    const float* __restrict__ a_src,
                                                    const float* __restrict__ a_dst,
                                                    float* __restrict__ as,
                                                    float* __restrict__ ad) {
    int n = blockIdx.x * 256 + threadIdx.x;
    if (n >= NTOT) return;
    float s = 0.f, d = 0.f;
    #pragma unroll
    for (int f = 0; f < HDIM; ++f) {
        float h = Hm[(size_t)n * HDIM + f];
        s += h * a_src[f]; d += h * a_dst[f];
    }
    as[n] = s; ad[n] = d;
}

// ---------- 5) init segment buffers + output<-bias ----------
__global__ __launch_bounds__(256) void init_kernel(unsigned* __restrict__ menc,
                                                   float* __restrict__ den,
                                                   float* __restrict__ outb,
                                                   const float* __restrict__ bias) {
    int i = blockIdx.x * 256 + threadIdx.x;          // NTOT*HDIM threads
    if (i < NTOT) { menc[i] = ENC_NEG_INF; den[i] = 0.f; }
    if (i < NTOT * HDIM) outb[i] = bias[i & (HDIM - 1)];
}

// ---------- 6) edge pass 1: leaky-relu logits + segment max over dst ----------
__global__ __launch_bounds__(256) void edge_max_kernel(const int* __restrict__ knn,
                                                       const float* __restrict__ as,
                                                       const float* __restrict__ ad,
                                                       unsigned* __restrict__ menc,
                                                       float* __restrict__ ev) {
    int e = blockIdx.x * 256 + threadIdx.x;
    if (e >= NETOT) return;
    int i, j;
    if (e < NEDGE) { i = e >> 4; j = knn[e]; } else { i = j = e - NEDGE; }
    float v = as[i] + ad[j];
    v = (v > 0.f) ? v : NEG_SLOPE * v;
    ev[e] = v;
    atomicMax(&menc[j], fenc(v));
}

// ---------- 7) edge pass 2: exp + segment sum ----------
__global__ __launch_bounds__(256) void edge_sum_kernel(const int* __restrict__ knn,
                                                       const unsigned* __restrict__ menc,
                                                       float* __restrict__ ev,
                                                       float* __restrict__ den) {
    int e = blockIdx.x * 256 + threadIdx.x;
    if (e >= NETOT) return;
    int j = (e < NEDGE) ? knn[e] : (e - NEDGE);
    float ex = expf(ev[e] - fdec(menc[j]));
    ev[e] = ex;
    atomicAdd(&den[j], ex);
}

// ---------- 8) edge pass 3: weighted feature scatter ----------
__global__ __launch_bounds__(256) void edge_agg_kernel(const int* __restrict__ knn,
                                                       const float* __restrict__ ev,
                                                       const float* __restrict__ den,
                                                       const float* __restrict__ Hm,
                                                       float* __restrict__ outb) {
    long long t = (long long)blockIdx.x * 256 + threadIdx.x;   // NETOT*16 threads
    int e = (int)(t >> 4);
    if (e >= NETOT) return;
    int f0 = (int)(t & 15) << 2;                               // 4 features each
    int i, j;
    if (e < NEDGE) { i = e >> 4; j = knn[e]; } else { i = j = e - NEDGE; }
    float alpha = ev[e] / (den[j] + 1e-16f);
    #pragma unroll
    for (int q = 0; q < 4; ++q)
        atomicAdd(&outb[(size_t)j * HDIM + f0 + q],
                  alpha * Hm[(size_t)i * HDIM + f0 + q]);
}

// ---------- 9) ELU (layer-1 activation, in place) ----------
__global__ __launch_bounds__(256) void elu_kernel(float* __restrict__ buf) {
    int i = blockIdx.x * 256 + threadIdx.x;
    if (i >= NTOT * HDIM) return;
    float x = buf[i];
    buf[i] = (x > 0.f) ? x : (expf(x) - 1.0f);
}

extern "C" void kernel_launch(void* const* d_in, const int* in_sizes, int n_in,
                              void* d_out, int out_size, void* d_ws, size_t ws_size,
                              hipStream_t stream) {
    (void)in_sizes; (void)n_in; (void)out_size; (void)ws_size;
    const float* x      = (const float*)d_in[0];
    const float* W1     = (const float*)d_in[1];
    const float* a_src1 = (const float*)d_in[2];
    const float* a_dst1 = (const float*)d_in[3];
    const float* b1     = (const float*)d_in[4];
    const float* W2     = (const float*)d_in[5];
    const float* a_src2 = (const float*)d_in[6];
    const float* a_dst2 = (const float*)d_in[7];
    const float* b2     = (const float*)d_in[8];
    float* out = (float*)d_out;

    float* ws = (float*)d_ws;
    float*    nodes = ws;                              // 524288
    float*    sq    = ws + 524288;                     // 16384
    int*      knnb  = (int*)(ws + 540672);             // 262144 ints
    float*    Hm    = ws + 802816;                     // 1048576
    float*    as    = ws + 1851392;                    // 16384
    float*    ad    = ws + 1867776;                    // 16384
    unsigned* menc  = (unsigned*)(ws + 1884160);       // 16384
    float*    den   = ws + 1900544;                    // 16384
    float*    ev    = ws + 1916928;                    // 278528
    float*    agg   = ws + 2195456;                    // 1048576

    const int EB = (NETOT + 255) / 256;                // edge-pass blocks
    const int AB = (int)(((long long)NETOT * 16 + 255) / 256);
    const size_t knn_lds = (size_t)(16 * NMPB + 16 * CDIM + 16) * sizeof(float);

    prep_kernel<<<NTOT / 256, 256, 0, stream>>>(x, nodes, sq);
    knn_kernel<<<BB * (NMPB / 16), 512, knn_lds, stream>>>(nodes, sq, knnb);

    // ---- layer 1 ----
    feat_kernel<CDIM><<<NTOT / 16, 128, 0, stream>>>(nodes, W1, Hm);
    alpha_kernel<<<NTOT / 256, 256, 0, stream>>>(Hm, a_src1, a_dst1, as, ad);
    init_kernel<<<NTOT * HDIM / 256, 256, 0, stream>>>(menc, den, agg, b1);
    edge_max_kernel<<<EB, 256, 0, stream>>>(knnb, as, ad, menc, ev);
    edge_sum_kernel<<<EB, 256, 0, stream>>>(knnb, menc, ev, den);
    edge_agg_kernel<<<AB, 256, 0, stream>>>(knnb, ev, den, Hm, agg);
    elu_kernel<<<NTOT * HDIM / 256, 256, 0, stream>>>(agg);

    // ---- layer 2 ----
    feat_kernel<HDIM><<<NTOT / 16, 128, 0, stream>>>(agg, W2, Hm);
    alpha_kernel<<<NTOT / 256, 256, 0, stream>>>(Hm, a_src2, a_dst2, as, ad);
    init_kernel<<<NTOT * HDIM / 256, 256, 0, stream>>>(menc, den, out, b2);
    edge_max_kernel<<<EB, 256, 0, stream>>>(knnb, as, ad, menc, ev);
    edge_sum_kernel<<<EB, 256, 0, stream>>>(knnb, menc, ev, den);
    edge_agg_kernel<<<AB, 256, 0, stream>>>(knnb, ev, den, Hm, out);
}